// roipooling_38577396253392
// MI455X (gfx1250) — compile-verified
//
#include <hip/hip_runtime.h>
#include <cstdint>

// ROI max pooling for MI455X (gfx1250).
// Strategy: one workgroup per ROI. The ROI's clipped feature-map region
// (rw x rh, up to 50x50) for a chunk of CC channels is DMA'd into LDS with a
// single TENSOR_LOAD_TO_LDS (3-D tile: x=cols, y=rows stride W, z=channels
// stride H*W), double-buffered and synchronized with s_wait_tensorcnt.
// Pooling then runs entirely out of LDS.

#define ROIP 7
#define CC   4      // channels per TDM chunk
#define MAXR 50     // max region extent (H = W = 50)

typedef unsigned int uint32x4 __attribute__((ext_vector_type(4)));
typedef int          int32x8  __attribute__((ext_vector_type(8)));
typedef int          int32x4  __attribute__((ext_vector_type(4)));

__device__ __forceinline__ int clampi(int v, int lo, int hi) {
    return v < lo ? lo : (v > hi ? hi : v);
}

// Issue one TDM load: global (gaddr, 3-D tile rw x rh x cc, strides in
// elements) -> LDS at byte offset lds_addr, packed [cc][rh][rw] f32.
__device__ __forceinline__ void tdm_load_tile(uint64_t gaddr, uint32_t lds_addr,
                                              int rw, int rh, int cc,
                                              uint64_t row_stride,
                                              uint64_t chan_stride)
{
    // ---- D# group 0 (128b): count | lds_addr | global_addr[56:0] | type=2
    uint32x4 g0;
    g0[0] = 1u;                                           // count=1, no gather
    g0[1] = lds_addr;                                     // LDS byte address
    g0[2] = (uint32_t)(gaddr & 0xffffffffull);            // global_addr[31:0]
    g0[3] = (uint32_t)((gaddr >> 32) & 0x01ffffffull)     // global_addr[56:32]
          | (2u << 30);                                   // type = 2 ("image")

    const uint32_t urw = (uint32_t)rw, urh = (uint32_t)rh;

    // ---- D# group 1 (256b)
    int32x8 g1;
    g1[0] = (int)(2u << 16);                              // data_size=2 (4B), mask=0
    g1[1] = (int)((urw & 0xffffu) << 16);                 // tensor_dim0[15:0]
    g1[2] = (int)((urw >> 16) | ((urh & 0xffffu) << 16)); // dim0[31:16] | dim1[15:0]
    g1[3] = (int)((urh >> 16) | ((urw & 0xffffu) << 16)); // dim1[31:16] | tile_dim0
    g1[4] = (int)((urh & 0xffffu) | (((uint32_t)cc) << 16)); // tile_dim1 | tile_dim2
    g1[5] = (int)(uint32_t)(row_stride & 0xffffffffull);  // dim0_stride[31:0]
    g1[6] = (int)(uint32_t)(((row_stride >> 32) & 0xffffull)
                            | ((chan_stride & 0xffffull) << 16)); // d0s[47:32]|d1s[15:0]
    g1[7] = (int)(uint32_t)((chan_stride >> 16) & 0xffffffffull); // d1s[47:16]

    // ---- D# group 2 (128b): tensor_dim2 = cc, tensor_dim3 = 1, tile_dim3 = 0
    int32x4 g2; g2[0] = cc; g2[1] = 1; g2[2] = 0; g2[3] = 0;
    // ---- D# group 3 (128b): unused
    int32x4 g3; g3[0] = 0; g3[1] = 0; g3[2] = 0; g3[3] = 0;

#if defined(__clang_major__) && __clang_major__ >= 23
    // amdgpu-toolchain (clang-23 / therock headers): 6-arg form
    int32x8 z8 = {0, 0, 0, 0, 0, 0, 0, 0};
    __builtin_amdgcn_tensor_load_to_lds(g0, g1, g2, g3, z8, 0);
#else
    // ROCm 7.2 (clang-22): 5-arg form
    __builtin_amdgcn_tensor_load_to_lds(g0, g1, g2, g3, 0);
#endif
}

__global__ __launch_bounds__(256)
void roipool_tdm_kernel(const float* __restrict__ feat,
                        const float* __restrict__ rois,
                        const float* __restrict__ scaling,
                        float* __restrict__ out,
                        int C, int H, int W)
{
    __shared__ float lds[2][CC * MAXR * MAXR];   // 2 x 40000 B, double buffer

    const int   n    = blockIdx.x;
    const float scal = scaling[0];

    const float r0 = rois[n * 5 + 0];
    const float r1 = rois[n * 5 + 1];
    const float r2 = rois[n * 5 + 2];
    const float r3 = rois[n * 5 + 3];
    const float r4 = rois[n * 5 + 4];

    const int b  = (int)r0;
    // jnp.round == round-to-nearest-even == rintf in default mode
    const int x1 = (int)rintf(r1 * scal);
    const int y1 = (int)rintf(r2 * scal);
    const int x2 = (int)rintf(r3 * scal);
    const int y2 = (int)rintf(r4 * scal);
    const int roi_w = max(x2 - x1 + 1, 1);
    const int roi_h = max(y2 - y1 + 1, 1);
    const float bsw = (float)roi_w / (float)ROIP;
    const float bsh = (float)roi_h / (float)ROIP;

    // Union of all bins (bin bounds are monotone in pw/ph): region to stage.
    const int w0 = clampi((int)floorf(0.0f * bsw) + x1, 0, W);
    const int wN = clampi((int)ceilf((float)ROIP * bsw) + x1, 0, W);
    const int h0 = clampi((int)floorf(0.0f * bsh) + y1, 0, H);
    const int hN = clampi((int)ceilf((float)ROIP * bsh) + y1, 0, H);
    const int rw = wN - w0;
    const int rh = hN - h0;
    const bool has_region = (rw > 0) && (rh > 0);

    // Per-thread bin assignment: t -> (channel-in-chunk, ph, pw)
    const int  t      = threadIdx.x;
    const int  cl     = t / (ROIP * ROIP);
    const int  bin    = t % (ROIP * ROIP);
    const int  ph     = bin / ROIP;
    const int  pw     = bin % ROIP;
    const bool active = (t < CC * ROIP * ROIP);

    // Mirror reference fp32 math exactly.
    const int wsB = clampi((int)floorf((float)pw * bsw) + x1, 0, W);
    const int weB = clampi((int)ceilf(((float)pw + 1.0f) * bsw) + x1, 0, W);
    const int hsB = clampi((int)floorf((float)ph * bsh) + y1, 0, H);
    const int heB = clampi((int)ceilf(((float)ph + 1.0f) * bsh) + y1, 0, H);
    const bool empty = (heB <= hsB) || (weB <= wsB);

    const uint64_t gbase       = (uint64_t)(uintptr_t)feat;
    const uint64_t chan_stride = (uint64_t)H * (uint64_t)W;   // elements
    const uint32_t lds_lo      = (uint32_t)(uintptr_t)&lds[0][0];
    const uint32_t lds_hi      = (uint32_t)(uintptr_t)&lds[1][0];
    const bool     leader      = (threadIdx.x < 32);          // wave 0 (wave32)
    const int      nchunk      = C / CC;

    if (leader && has_region) {
        uint64_t ga = gbase +
            4ull * ((((uint64_t)(b * C)) * (uint64_t)H + (uint64_t)h0)
                    * (uint64_t)W + (uint64_t)w0);
        tdm_load_tile(ga, lds_lo, rw, rh, CC, (uint64_t)W, chan_stride);
    }

    for (int ch = 0; ch < nchunk; ++ch) {
        if (leader && has_region) {
            if (ch + 1 < nchunk) {
                uint64_t ga = gbase +
                    4ull * ((((uint64_t)(b * C + (ch + 1) * CC)) * (uint64_t)H
                             + (uint64_t)h0) * (uint64_t)W + (uint64_t)w0);
                tdm_load_tile(ga, ((ch + 1) & 1) ? lds_hi : lds_lo,
                              rw, rh, CC, (uint64_t)W, chan_stride);
                __builtin_amdgcn_s_wait_tensorcnt(1);  // chunk ch complete (in-order)
            } else {
                __builtin_amdgcn_s_wait_tensorcnt(0);  // last chunk complete
            }
        }
        __syncthreads();   // chunk ch visible in LDS to all waves

        if (active) {
            const float* buf = (ch & 1) ? &lds[1][0] : &lds[0][0];
            float m = -3.402823466e38f;
            for (int h = hsB; h < heB; ++h) {
                const float* row = buf + ((size_t)cl * (size_t)rh + (size_t)(h - h0))
                                         * (size_t)rw - (size_t)w0;
                for (int w = wsB; w < weB; ++w)
                    m = fmaxf(m, row[w]);
            }
            const int c = ch * CC + cl;
            out[((size_t)n * (size_t)C + (size_t)c) * (ROIP * ROIP) + bin] =
                empty ? 0.0f : m;
        }
        __syncthreads();   // buffer free before wave 0 overwrites it next iter
    }
}

extern "C" void kernel_launch(void* const* d_in, const int* in_sizes, int n_in,
                              void* d_out, int out_size, void* d_ws, size_t ws_size,
                              hipStream_t stream) {
    (void)n_in; (void)out_size; (void)d_ws; (void)ws_size;
    const float* feat = (const float*)d_in[0];   // [B, C, H, W] f32
    const float* rois = (const float*)d_in[1];   // [N, 5] f32
    const float* scal = (const float*)d_in[2];   // scalar f32
    float*       out  = (float*)d_out;           // [N, C, 7, 7] f32

    const int C = 256, H = 50, W = 50;           // fixed by reference setup
    const int N = in_sizes[1] / 5;

    dim3 grid((unsigned)N), block(256);
    roipool_tdm_kernel<<<grid, block, 0, stream>>>(feat, rois, scal, out, C, H, W);
}